// Model_87333864997440
// MI455X (gfx1250) — compile-verified
//
#include <hip/hip_runtime.h>
#include <hip/hip_bf16.h>

typedef int   __attribute__((ext_vector_type(4))) i32x4;
typedef float __attribute__((ext_vector_type(4))) f32x4;

// ---------------------------------------------------------------------------
// Kernel 1: bulk copy-convert of output_bin_counts (int32 -> float32).
// 262 MB stream, larger than the 192 MB L2 and never re-read, so use
// non-temporal (TH_NT) B128 loads/stores to stream straight through the
// cache hierarchy without rinsing useful lines.
// ---------------------------------------------------------------------------
__global__ __launch_bounds__(256)
void copy_bins_nt(const int* __restrict__ bins_in,
                  float* __restrict__ bins_out,
                  long long n4) {
    long long idx = (long long)blockIdx.x * blockDim.x + threadIdx.x;
    if (idx >= n4) return;
    const i32x4* __restrict__ src = (const i32x4*)bins_in;
    f32x4* __restrict__ dst       = (f32x4*)bins_out;
    i32x4 v = __builtin_nontemporal_load(&src[idx]);   // global_load_b128 th:NT
    f32x4 f;
    f.x = (float)v.x;
    f.y = (float)v.y;
    f.z = (float)v.z;
    f.w = (float)v.w;
    __builtin_nontemporal_store(f, &dst[idx]);         // global_store_b128 th:NT
}

// ---------------------------------------------------------------------------
// Kernel 2: single block. Phase A copies the two small state vectors (and any
// non-multiple-of-4 tail of bin_counts); phase B applies the per-request
// scatter updates. idx_mapping rows are unique, so one thread per request can
// do plain sequential RMW — duplicate tokens inside a request are handled
// in-order by the owning thread. No atomics required.
// ---------------------------------------------------------------------------
__global__ void state_update(const int* __restrict__ idx_mapping,
                             const int* __restrict__ nct_in,
                             const int* __restrict__ last_in,
                             const int* __restrict__ bins_in,
                             const int* __restrict__ sampled,
                             const int* __restrict__ num_sampled,
                             const int* __restrict__ num_rejected,
                             const int* __restrict__ qsl,
                             float* __restrict__ nct_out,
                             float* __restrict__ last_out,
                             float* __restrict__ bins_out,
                             int num_reqs, int max_reqs, int vocab, int s1,
                             long long tail_start, long long bin_total) {
    const int t = threadIdx.x;
    const int nthreads = blockDim.x;

    // --- Phase A: materialize unchanged state into d_out ---
    for (int j = t; j < max_reqs; j += nthreads) {
        nct_out[j]  = (float)nct_in[j];
        last_out[j] = (float)last_in[j];
    }
    for (long long j = tail_start + t; j < bin_total; j += nthreads) {
        bins_out[j] = (float)bins_in[j];
    }
    __syncthreads();

    // --- Phase B: per-request scatter updates ---
    for (int r = t; r < num_reqs; r += nthreads) {
        const int s    = idx_mapping[r];
        const int ns   = num_sampled[r];
        const int qlen = qsl[r + 1] - qsl[r];

        // num_computed_tokens[s] += query_len - num_rejected
        nct_out[s] += (float)(qlen - num_rejected[r]);

        if (ns > 0) {
            int li = ns - 1;
            if (li >= s1) li = s1 - 1;
            last_out[s] = (float)sampled[r * s1 + li];

            const long long base = (long long)s * (long long)vocab;
            const int lim = ns < s1 ? ns : s1;
            for (int i = 0; i < lim; ++i) {
                const int tok = sampled[r * s1 + i];
                bins_out[base + tok] += 1.0f;   // unique row -> race-free RMW
            }
        }
    }
}

extern "C" void kernel_launch(void* const* d_in, const int* in_sizes, int n_in,
                              void* d_out, int out_size, void* d_ws, size_t ws_size,
                              hipStream_t stream) {
    const int* idx_mapping  = (const int*)d_in[0];
    const int* nct_in       = (const int*)d_in[1];
    const int* last_in      = (const int*)d_in[2];
    const int* bins_in      = (const int*)d_in[3];
    const int* sampled      = (const int*)d_in[4];
    const int* num_sampled  = (const int*)d_in[5];
    const int* num_rejected = (const int*)d_in[6];
    const int* qsl          = (const int*)d_in[7];

    const int num_reqs        = in_sizes[0];             // 256
    const int max_reqs        = in_sizes[1];             // 512
    const long long bin_total = (long long)in_sizes[3];  // 512*128000
    const int vocab           = (int)(bin_total / max_reqs);
    const int s1              = in_sizes[4] / num_reqs;  // 9

    float* nct_out  = (float*)d_out;
    float* last_out = nct_out + max_reqs;
    float* bins_out = last_out + max_reqs;

    // Bulk stream: 4 int32 per thread, 256-thread (8-wave) blocks.
    const long long n4 = bin_total / 4;
    if (n4 > 0) {
        const int threads = 256;
        const long long blocks = (n4 + threads - 1) / threads;
        copy_bins_nt<<<dim3((unsigned)blocks), dim3(threads), 0, stream>>>(
            bins_in, bins_out, n4);
    }

    // Small-state + scatter kernel: one block sized to cover max_reqs.
    int T = max_reqs > num_reqs ? max_reqs : num_reqs;
    T = ((T + 31) / 32) * 32;
    if (T > 1024) T = 1024;
    state_update<<<1, dim3(T), 0, stream>>>(
        idx_mapping, nct_in, last_in, bins_in, sampled, num_sampled,
        num_rejected, qsl, nct_out, last_out, bins_out,
        num_reqs, max_reqs, vocab, s1, n4 * 4, bin_total);
}